// GaussianKernel_10771777978937
// MI455X (gfx1250) — compile-verified
//
#include <hip/hip_runtime.h>
#include <cmath>

#ifndef M_PI
#define M_PI 3.14159265358979323846
#endif

typedef __attribute__((ext_vector_type(2))) float v2f;
typedef __attribute__((ext_vector_type(8))) float v8f;

#define NQ 2048
#define ND 2048
#define DIMK 128
#define KSTEPS (DIMK / 4)                 // 32 wmma_f32_16x16x4 per 16x16 tile
#define COLSPLIT 8                        // column splits for occupancy
#define ROWTILES (NQ / 16)                // 128
#define COLTILES ((ND / 16) / COLSPLIT)   // 16 tiles of 16 cols per wave

// ---------------------------------------------------------------------------
// Kernel 1: half squared norms. One wave per row; lane loads float4, 5-step
// shfl reduction. hn[0..ND) = 0.5*||x_j||^2 ; hn[ND..ND+NQ) = 0.5*||y_i||^2.
// ---------------------------------------------------------------------------
__global__ void __launch_bounds__(128) kde_halfnorms(
    const float* __restrict__ y, const float* __restrict__ x,
    float* __restrict__ hn) {
  const int wave = (blockIdx.x * blockDim.x + threadIdx.x) >> 5;
  const int lane = threadIdx.x & 31;
  const float* src = (wave < ND) ? x : y;
  const int row = (wave < ND) ? wave : (wave - ND);
  const float4 v = *(const float4*)(src + (size_t)row * DIMK + lane * 4);
  float s = v.x * v.x + v.y * v.y + v.z * v.z + v.w * v.w;
  #pragma unroll
  for (int d = 16; d >= 1; d >>= 1) s += __shfl_xor(s, d, 32);
  if (lane == 0) hn[wave] = 0.5f * s;
}

// ---------------------------------------------------------------------------
// Kernel 2: fused  Y·X^T (fp32 WMMA)  +  online base-2 logsumexp.
// One wave (32 threads) per {16-row stripe, 256-column split}.
//
// A-fragment layout (V_WMMA_F32_16X16X4_F32, ISA 7.12.2): lane holds row
// (lane&15); VGPR0/1 hold K = kb, kb+1 with kb = 2*(lane>>4). B mirrors A
// with N = lane&15. C/D: col = lane&15, row = v + 8*(lane>>4) for VGPR v.
//
// B fragments are staged through a 2-deep register ring in chunks of 8 so
// 8 global_load_b64 stay in flight behind the WMMAs (partial loadcnt waits),
// and the K accumulation is split by parity across two accumulators (c0/c1)
// so consecutive WMMAs are independent on the matrix pipe.
// ---------------------------------------------------------------------------
__global__ void __launch_bounds__(32) kde_wmma_main(
    const float* __restrict__ y, const float* __restrict__ x,
    const float* __restrict__ hxn, const float* __restrict__ hyn,
    float2* __restrict__ partial, float scale /* log2(e)/h^2 */) {
  const int lane = threadIdx.x;
  const int rT = blockIdx.x & (ROWTILES - 1);
  const int split = blockIdx.x >> 7;        // blockIdx.x / ROWTILES
  const int r = lane & 15;
  const int kb = (lane >> 4) << 1;          // 0 or 2
  const int rowOff = (lane >> 4) << 3;      // C-tile row offset: 0 or 8

  // Preload the A fragment for this query stripe once (reused 16x).
  v2f a[KSTEPS];
  const float* yrow = y + (size_t)(rT * 16 + r) * DIMK + kb;
  #pragma unroll
  for (int ks = 0; ks < KSTEPS; ++ks) a[ks] = *(const v2f*)(yrow + ks * 4);

  float hy[8];
  #pragma unroll
  for (int v = 0; v < 8; ++v) hy[v] = hyn[rT * 16 + rowOff + v];

  float m[8], s[8];
  #pragma unroll
  for (int v = 0; v < 8; ++v) { m[v] = -3.0e38f; s[v] = 0.0f; }

  for (int ct = 0; ct < COLTILES; ++ct) {
    const int col = (split * COLTILES + ct) * 16 + r;
    const float* xcol = x + (size_t)col * DIMK + kb;
    if (ct + 1 < COLTILES)                       // stream next tile into cache
      __builtin_prefetch(xcol + 16 * DIMK, 0, 1);
    const float hx = hxn[col];

    v8f c0 = {0.f, 0.f, 0.f, 0.f, 0.f, 0.f, 0.f, 0.f};
    v8f c1 = {0.f, 0.f, 0.f, 0.f, 0.f, 0.f, 0.f, 0.f};

    // 2-deep register ring of 8 B fragments (8 loads in flight).
    v2f b[2][8];
    #pragma unroll
    for (int j = 0; j < 8; ++j) b[0][j] = *(const v2f*)(xcol + j * 4);

    #pragma unroll
    for (int kc = 0; kc < 4; ++kc) {
      const int cur = kc & 1;
      if (kc < 3) {
        #pragma unroll
        for (int j = 0; j < 8; ++j)
          b[cur ^ 1][j] = *(const v2f*)(xcol + (kc + 1) * 32 + j * 4);
      }
      #pragma unroll
      for (int j = 0; j < 8; ++j) {
        const int ks = kc * 8 + j;
        if (j & 1) {
          c1 = __builtin_amdgcn_wmma_f32_16x16x4_f32(
              false, a[ks], false, b[cur][j], (short)0, c1, false, false);
        } else {
          c0 = __builtin_amdgcn_wmma_f32_16x16x4_f32(
              false, a[ks], false, b[cur][j], (short)0, c0, false, false);
        }
      }
    }

    // Online logsumexp update, base-2 domain (v_exp_f32 native).
    #pragma unroll
    for (int v = 0; v < 8; ++v) {
      const float t = (c0[v] + c1[v] - hy[v] - hx) * scale;
      const float nm = fmaxf(m[v], t);
      s[v] = s[v] * __builtin_amdgcn_exp2f(m[v] - nm) +
             __builtin_amdgcn_exp2f(t - nm);
      m[v] = nm;
    }
  }

  // Merge the 16 column phases per row (stays inside each 16-lane half).
  #pragma unroll
  for (int v = 0; v < 8; ++v) {
    #pragma unroll
    for (int d = 1; d < 16; d <<= 1) {
      const float om = __shfl_xor(m[v], d, 32);
      const float os = __shfl_xor(s[v], d, 32);
      const float nm = fmaxf(m[v], om);
      s[v] = s[v] * __builtin_amdgcn_exp2f(m[v] - nm) +
             os * __builtin_amdgcn_exp2f(om - nm);
      m[v] = nm;
    }
    if (r == v) {  // lanes 0..7 write rows 0..7, lanes 16..23 write rows 8..15
      const int grow = rT * 16 + rowOff + v;
      partial[split * NQ + grow] = make_float2(m[v], s[v]);
    }
  }
}

// ---------------------------------------------------------------------------
// Kernel 3: merge the 8 column-split partials, convert base-2 -> ln, apply -Z.
// ---------------------------------------------------------------------------
__global__ void __launch_bounds__(256) kde_finalize(
    const float2* __restrict__ partial, float* __restrict__ out, float Z) {
  const int i = blockIdx.x * blockDim.x + threadIdx.x;
  if (i >= NQ) return;
  float m = -3.0e38f, s = 0.0f;
  #pragma unroll
  for (int p = 0; p < COLSPLIT; ++p) {
    const float2 v = partial[p * NQ + i];
    const float nm = fmaxf(m, v.x);
    s = s * __builtin_amdgcn_exp2f(m - nm) +
        v.y * __builtin_amdgcn_exp2f(v.x - nm);
    m = nm;
  }
  out[i] = 0.693147180559945309f * (m + log2f(s)) - Z;
}

// ---------------------------------------------------------------------------
extern "C" void kernel_launch(void* const* d_in, const int* in_sizes, int n_in,
                              void* d_out, int out_size, void* d_ws,
                              size_t ws_size, hipStream_t stream) {
  const float* y = (const float*)d_in[0];
  const float* x = (const float*)d_in[1];
  float* out = (float*)d_out;

  float* ws = (float*)d_ws;
  float* hxn = ws;                    // [ND]
  float* hyn = ws + ND;               // [NQ]
  float2* partial = (float2*)(ws + ND + NQ);  // [COLSPLIT * NQ] float2

  const double bw = 0.1;
  const float Z = (float)(0.5 * DIMK * std::log(2.0 * M_PI) +
                          DIMK * std::log(bw) + std::log((double)ND));
  const float scale = (float)(1.4426950408889634 / (bw * bw));  // log2e / h^2

  kde_halfnorms<<<(ND + NQ) / 4, 128, 0, stream>>>(y, x, ws);
  kde_wmma_main<<<ROWTILES * COLSPLIT, 32, 0, stream>>>(y, x, hxn, hyn,
                                                        partial, scale);
  kde_finalize<<<(NQ + 255) / 256, 256, 0, stream>>>(partial, out, Z);
}